// Model_5927054869109
// MI455X (gfx1250) — compile-verified
//
#include <hip/hip_runtime.h>
#include <hip/hip_bf16.h>

typedef __attribute__((ext_vector_type(2))) float v2f;
typedef __attribute__((ext_vector_type(8))) float v8f;

#define DIN  256
#define DOUT 128

// ---------------------------------------------------------------------------
// Zero-fill workspace accumulators
// ---------------------------------------------------------------------------
__global__ void zero_f32(float* __restrict__ p, size_t n) {
    size_t i = (size_t)blockIdx.x * blockDim.x + threadIdx.x;
    if (i < n) p[i] = 0.0f;
}

// ---------------------------------------------------------------------------
// GEMM + bias: out[M,128] = X[M,256] @ W[256,128] + b
// One wave computes a 16-row x 128-col strip using V_WMMA_F32_16X16X4_F32.
// Block = 128 threads = 4 waves = 64 rows. M is a multiple of 16 (200000).
//
// Fragment layouts (wave32, ISA 7.12.2):
//   A 16x4 f32 : lanes 0-15 hold (K=0,K=1) for row M=lane; lanes 16-31 (K=2,K=3)
//   B 4x16 f32 : lanes 0-15 hold (K=0,K=1) for col N=lane; lanes 16-31 (K=2,K=3)
//   C/D 16x16  : VGPR r -> lanes 0-15: (M=r, N=lane); lanes 16-31: (M=8+r, N=lane-16)
// ---------------------------------------------------------------------------
__global__ __launch_bounds__(128) void gemm_bias_wmma(
    const float* __restrict__ X, const float* __restrict__ W,
    const float* __restrict__ bias, float* __restrict__ out, int M)
{
    const int lane    = threadIdx.x & 31;
    const int wave    = threadIdx.x >> 5;
    const int m0      = blockIdx.x * 64 + wave * 16;
    if (m0 >= M) return;                       // uniform per wave
    const int halfsel = lane >> 4;             // 0: lanes 0-15, 1: lanes 16-31
    const int l16     = lane & 15;
    const int kbase   = halfsel * 2;           // upper half-wave covers K+2,K+3

    v8f acc[8];
    v8f zz = {};
    #pragma unroll
    for (int t = 0; t < 8; ++t) acc[t] = zz;

    const float* xrow = X + (size_t)(m0 + l16) * DIN;

    for (int k = 0; k < DIN; k += 4) {
        // A fragment: two consecutive K values for this lane's row (8B load)
        const v2f a = *(const v2f*)(xrow + k + kbase);
        #pragma unroll
        for (int t = 0; t < 8; ++t) {
            const int col = t * 16 + l16;
            v2f b;
            b.x = W[(size_t)(k + kbase)     * DOUT + col];
            b.y = W[(size_t)(k + kbase + 1) * DOUT + col];
            acc[t] = __builtin_amdgcn_wmma_f32_16x16x4_f32(
                /*neg_a=*/false, a, /*neg_b=*/false, b,
                /*c_mod=*/(short)0, acc[t], /*reuse_a=*/false, /*reuse_b=*/false);
        }
    }

    #pragma unroll
    for (int t = 0; t < 8; ++t) {
        const int col = t * 16 + l16;
        const float bv = bias[col];
        #pragma unroll
        for (int r = 0; r < 8; ++r) {
            const int row = m0 + r + halfsel * 8;
            out[(size_t)row * DOUT + col] = acc[t][r] + bv;
        }
    }
}

// ---------------------------------------------------------------------------
// Scatter-accumulate: sums[dst[e]] += msgs[src[e]], cnt[dst[e]] += 1
// One wave per edge; 128 feats = 4 floats/lane (float4 load, 4 f32 atomics).
// ---------------------------------------------------------------------------
__global__ __launch_bounds__(256) void scatter_accum(
    const float* __restrict__ msgs, const int* __restrict__ src,
    const int* __restrict__ dst, float* __restrict__ sums,
    float* __restrict__ cnt, int nE)
{
    const int e = blockIdx.x * 8 + (threadIdx.x >> 5);
    if (e >= nE) return;
    const int lane = threadIdx.x & 31;
    const int s = src[e];
    const int d = dst[e];
    const float4 v = *(const float4*)(msgs + (size_t)s * DOUT + lane * 4);
    float* base = sums + (size_t)d * DOUT + lane * 4;
    atomicAdd(base + 0, v.x);
    atomicAdd(base + 1, v.y);
    atomicAdd(base + 2, v.z);
    atomicAdd(base + 3, v.w);
    if (lane == 0) atomicAdd(cnt + d, 1.0f);
}

// ---------------------------------------------------------------------------
// In-place mean: sums[i,f] /= max(cnt[i], 1)
// ---------------------------------------------------------------------------
__global__ void finalize_mean(float* __restrict__ sums,
                              const float* __restrict__ cnt, size_t nNodes)
{
    size_t i = (size_t)blockIdx.x * blockDim.x + threadIdx.x;
    size_t total = nNodes * DOUT;
    if (i >= total) return;
    float c = cnt[i >> 7];
    sums[i] = sums[i] / fmaxf(c, 1.0f);
}

// ---------------------------------------------------------------------------
// Per-edge 128-dim dot product: out[e] = dot(hu[src[e]], hi[dst[e]])
// One wave per edge; wave32 butterfly reduction.
// ---------------------------------------------------------------------------
__global__ __launch_bounds__(256) void edge_dot(
    const float* __restrict__ hu, const float* __restrict__ hi,
    const int* __restrict__ src, const int* __restrict__ dst,
    float* __restrict__ out, int nE)
{
    const int e = blockIdx.x * 8 + (threadIdx.x >> 5);
    if (e >= nE) return;
    const int lane = threadIdx.x & 31;
    const int u = src[e];
    const int it = dst[e];
    const float4 a = *(const float4*)(hu + (size_t)u  * DOUT + lane * 4);
    const float4 b = *(const float4*)(hi + (size_t)it * DOUT + lane * 4);
    float s = a.x * b.x + a.y * b.y + a.z * b.z + a.w * b.w;
    #pragma unroll
    for (int off = 16; off > 0; off >>= 1)
        s += __shfl_xor(s, off, 32);
    if (lane == 0) out[e] = s;
}

// ---------------------------------------------------------------------------
extern "C" void kernel_launch(void* const* d_in, const int* in_sizes, int n_in,
                              void* d_out, int out_size, void* d_ws, size_t ws_size,
                              hipStream_t stream)
{
    const float* x_user     = (const float*)d_in[0];
    const float* x_item     = (const float*)d_in[1];
    const float* W_clicks   = (const float*)d_in[2];
    const float* b_clicks   = (const float*)d_in[3];
    const float* W_rev      = (const float*)d_in[4];
    const float* b_rev      = (const float*)d_in[5];
    const int*   clicks_src = (const int*)d_in[6];
    const int*   clicks_dst = (const int*)d_in[7];
    const int*   rev_src    = (const int*)d_in[8];
    const int*   rev_dst    = (const int*)d_in[9];
    const int*   neg_src    = (const int*)d_in[10];
    const int*   neg_dst    = (const int*)d_in[11];

    const int NU = in_sizes[0] / DIN;
    const int NI = in_sizes[1] / DIN;
    const int E  = in_sizes[6];
    const int ER = in_sizes[8];
    const int EN = in_sizes[10];

    float* ws       = (float*)d_ws;
    float* Wh_u     = ws;                                  // [NU,128]
    float* Wh_i     = Wh_u     + (size_t)NU * DOUT;        // [NI,128]
    float* sum_item = Wh_i     + (size_t)NI * DOUT;        // [NI,128]
    float* sum_user = sum_item + (size_t)NI * DOUT;        // [NU,128]
    float* cnt_item = sum_user + (size_t)NU * DOUT;        // [NI]
    float* cnt_user = cnt_item + (size_t)NI;               // [NU]

    // 1) zero accumulators (sum_item..cnt_user are contiguous)
    size_t zn = (size_t)NI * DOUT + (size_t)NU * DOUT + (size_t)NI + (size_t)NU;
    zero_f32<<<(unsigned)((zn + 255) / 256), 256, 0, stream>>>(sum_item, zn);

    // 2) per-etype linear layers (WMMA f32)
    gemm_bias_wmma<<<(NU + 63) / 64, 128, 0, stream>>>(x_user, W_clicks, b_clicks, Wh_u, NU);
    gemm_bias_wmma<<<(NI + 63) / 64, 128, 0, stream>>>(x_item, W_rev, b_rev, Wh_i, NI);

    // 3) copy_u + mean aggregation (atomic scatter then divide)
    scatter_accum<<<(E  + 7) / 8, 256, 0, stream>>>(Wh_u, clicks_src, clicks_dst, sum_item, cnt_item, E);
    scatter_accum<<<(ER + 7) / 8, 256, 0, stream>>>(Wh_i, rev_src,    rev_dst,    sum_user, cnt_user, ER);
    finalize_mean<<<(unsigned)(((size_t)NI * DOUT + 255) / 256), 256, 0, stream>>>(sum_item, cnt_item, (size_t)NI);
    finalize_mean<<<(unsigned)(((size_t)NU * DOUT + 255) / 256), 256, 0, stream>>>(sum_user, cnt_user, (size_t)NU);

    // 4) dot-product predictor: pos then neg, concatenated in d_out
    float* out = (float*)d_out;
    edge_dot<<<(E  + 7) / 8, 256, 0, stream>>>(sum_user, sum_item, clicks_src, clicks_dst, out,     E);
    edge_dot<<<(EN + 7) / 8, 256, 0, stream>>>(sum_user, sum_item, neg_src,    neg_dst,    out + E, EN);
}